// TorchLFQ_26336739459235
// MI455X (gfx1250) — compile-verified
//
#include <hip/hip_runtime.h>
#include <hip/hip_bf16.h>

// LFQ entropy loss, D=20 bits, K=2^20 codes, x:[8,256,20] f32.
// Heavy term reformulated as 16 small f32 GEMMs ([512x256]@[256x1024]) fed to
// V_WMMA_F32_16X16X4_F32, with the -p*log(p+eps) reduction fused on the
// accumulator registers. Per-token entropy / commit loss are closed-form O(D).

#define LFQ_TEMP    0.005f
#define LFQ_INVT    200.0f          // 1/TEMP
#define LFQ_EPS     1e-10f
#define NTOK        2048            // 8*256
#define NLO         1024            // 2^10 low free bits
#define NHI         512             // 2^9  high free bits

typedef __attribute__((ext_vector_type(2))) float v2f;
typedef __attribute__((ext_vector_type(8))) float v8f;

// ---------------------------------------------------------------- init ------
__global__ void lfq_init(float* __restrict__ acc) {
    if (threadIdx.x < 4) acc[threadIdx.x] = 0.0f;
}

// ------------------------------------------------------- per-token setup ----
// Block = one token (512 threads). Produces:
//   q = sign(x)                         -> d_out[0..40959]
//   acc[0] += e_t (closed-form entropy), acc[2] += commit partial
//   Elo[t][a]  (a=0..1023): exp(sum over 10 low-bit dims of +-y)
//   W1[b][h][t], W2[b][h][t] (h=0..511): u1/u2 * exp(sum over 9 high-bit dims)
// Bit p of code k maps to dim j = 19-p, sign +1 iff bit set; dim 0 sign -1 in W1, +1 in W2.
__global__ void lfq_setup(const float* __restrict__ x,
                          float* __restrict__ qout,
                          float* __restrict__ elo,
                          float* __restrict__ w1,
                          float* __restrict__ w2,
                          float* __restrict__ acc) {
    const int t  = blockIdx.x;          // 0..2047
    const int b  = t >> 8;
    const int tr = t & 255;
    const int tid = threadIdx.x;        // 0..511

    __shared__ float y[20];
    __shared__ float su1, su2;

    if (tid < 20) {
        float xv = x[t * 20 + tid];
        y[tid] = xv * LFQ_INVT;
        qout[t * 20 + tid] = (xv > 0.0f) ? 1.0f : -1.0f;
    }
    __syncthreads();

    if (tid == 0) {
        float logC = 0.0f, T1 = 0.0f, commit = 0.0f;
        #pragma unroll
        for (int j = 0; j < 20; ++j) {
            float yj = y[j];
            float ay = fabsf(yj);
            logC += ay + log1pf(__expf(-2.0f * ay));   // log(2*cosh(y))
            if (j > 0) T1 += yj * tanhf(yj);
            float xv = yj * LFQ_TEMP;
            float d  = xv - ((xv > 0.0f) ? 1.0f : -1.0f);
            commit += d * d;
        }
        float y0  = y[0];
        T1 += -y0;
        float sig = 1.0f / (1.0f + __expf(-2.0f * y0));        // Ht/Ct
        float e_t = -(T1 - logC + 2.0f * y0 * sig);            // >= 0
        atomicAdd(&acc[0], e_t);
        atomicAdd(&acc[2], commit);
        su1 = __expf(-y0 - logC) * (1.0f / 256.0f);            // m-weight
        su2 = __expf( y0 - logC) * (1.0f / 256.0f);            // n-weight
    }
    __syncthreads();

    const float u1 = su1, u2 = su2;

    // Elo: low 10 bits -> dims 19..10
    for (int a = tid; a < NLO; a += 512) {
        float s = 0.0f;
        #pragma unroll
        for (int p = 0; p < 10; ++p)
            s += ((a >> p) & 1) ? y[19 - p] : -y[19 - p];
        elo[t * NLO + a] = __expf(s);
    }
    // Ehi (weighted): high 9 bits -> dims 9..1 ; layout [b][h][t] for A-loads
    {
        int h = tid;                    // 0..511
        float s = 0.0f;
        #pragma unroll
        for (int p = 0; p < 9; ++p)
            s += ((h >> p) & 1) ? y[9 - p] : -y[9 - p];
        float E = __expf(s);
        int idx = (b * NHI + h) * 256 + tr;
        w1[idx] = u1 * E;
        w2[idx] = u2 * E;
    }
}

// ------------------------------------------- WMMA GEMM + fused entropy ------
// grid = 8(b) * 32(h-tile) * 8(a-group) blocks; 256 thr = 8 waves; each wave
// owns one 16x16 (h,a) tile, accumulating P1,P2 over K=256 tokens in k=4 steps.
__global__ void lfq_gemm_entropy(const float* __restrict__ elo,
                                 const float* __restrict__ w1,
                                 const float* __restrict__ w2,
                                 float* __restrict__ acc) {
    const int blk  = blockIdx.x;
    const int ag   = blk & 7;
    const int ht   = (blk >> 3) & 31;
    const int b    = blk >> 8;
    const int wave = threadIdx.x >> 5;
    const int lane = threadIdx.x & 31;

    const int h0 = ht * 16;
    const int a0 = (ag * 8 + wave) * 16;
    const int mn = lane & 15;           // M for A, N for B
    const int kk = (lane >> 4) << 1;    // K-pair selector per ISA f32 layout

    const float* w1p = w1 + (b * NHI + h0 + mn) * 256 + kk;   // A1[m][k..k+1]
    const float* w2p = w2 + (b * NHI + h0 + mn) * 256 + kk;   // A2[m][k..k+1]
    const float* bp  = elo + (b * 256 + kk) * NLO + a0 + mn;  // B[k][n]

    v8f p1 = {};
    v8f p2 = {};
    for (int t0 = 0; t0 < 256; t0 += 4) {
        if (t0 + 4 < 256)
            __builtin_prefetch(bp + (t0 + 4) * NLO, 0, 1);    // global_prefetch_b8
        v2f A1 = *(const v2f*)(w1p + t0);
        v2f A2 = *(const v2f*)(w2p + t0);
        v2f B;
        B.x = bp[t0 * NLO];
        B.y = bp[t0 * NLO + NLO];
#if defined(__has_builtin) && __has_builtin(__builtin_amdgcn_wmma_f32_16x16x4_f32)
        p1 = __builtin_amdgcn_wmma_f32_16x16x4_f32(false, A1, false, B,
                                                   (short)0, p1, false, false);
        p2 = __builtin_amdgcn_wmma_f32_16x16x4_f32(false, A2, false, B,
                                                   (short)0, p2, false, false);
#else   // host pass / missing builtin: reference semantics (per-lane partial)
        p1[0] += A1.x * B.x + A1.y * B.y;
        p2[0] += A2.x * B.x + A2.y * B.y;
#endif
    }

    // fused -sum p*log(p+eps): argument >= 1e-10 (never denormal), so the
    // hardware v_log_f32 path (__logf) is safe and strips the precise-logf
    // denorm-rescale VALU ladder from this 8.4M-element epilogue.
    float local = 0.0f;
    #pragma unroll
    for (int i = 0; i < 8; ++i) {
        float a1 = p1[i]; local += a1 * __logf(a1 + LFQ_EPS);
        float a2 = p2[i]; local += a2 * __logf(a2 + LFQ_EPS);
    }
    for (int off = 16; off > 0; off >>= 1)
        local += __shfl_down(local, off, 32);
    if (lane == 0) atomicAdd(&acc[1], local);
}

// ----------------------------------------------------------- finalize -------
__global__ void lfq_finalize(const float* __restrict__ acc,
                             float* __restrict__ out) {
    if (threadIdx.x == 0) {
        float entro_mean = acc[0] * (1.0f / 2048.0f);   // mean_t e_t
        float mean_entro = -acc[1] * (1.0f / 8.0f);     // mean_b(-sum p log p)
        out[NTOK * 20 + 0] = entro_mean;
        out[NTOK * 20 + 1] = mean_entro;
        out[NTOK * 20 + 2] = entro_mean - mean_entro;   // ALPHA = 1
        out[NTOK * 20 + 3] = acc[2] * (1.0f / 40960.0f);
    }
}

// ------------------------------------------------------------- launch -------
extern "C" void kernel_launch(void* const* d_in, const int* in_sizes, int n_in,
                              void* d_out, int out_size, void* d_ws, size_t ws_size,
                              hipStream_t stream) {
    (void)in_sizes; (void)n_in; (void)out_size; (void)ws_size;
    const float* x = (const float*)d_in[0];
    float* out = (float*)d_out;

    // workspace layout (floats): acc[16] | Elo[2048*1024] | W1[8*512*256] | W2[...]
    float* acc = (float*)d_ws;
    float* elo = acc + 16;
    float* w1  = elo + (size_t)NTOK * NLO;
    float* w2  = w1  + (size_t)8 * NHI * 256;

    lfq_init<<<1, 32, 0, stream>>>(acc);
    lfq_setup<<<NTOK, 512, 0, stream>>>(x, out, elo, w1, w2, acc);
    lfq_gemm_entropy<<<8 * 32 * 8, 256, 0, stream>>>(elo, w1, w2, acc);
    lfq_finalize<<<1, 32, 0, stream>>>(acc, out);
}